// FMRConv_27376121545184
// MI455X (gfx1250) — compile-verified
//
#include <hip/hip_runtime.h>
#include <cmath>

#define BB 4
#define NN 4096
#define SS 1024
#define NSK 32
#define VV 8
#define CIN 64
#define COUT_ 128
#define C1_ 70
#define RAD2 0.04f
#define EPSB 1e-5f

typedef float v2f __attribute__((ext_vector_type(2)));
typedef float v4f __attribute__((ext_vector_type(4)));
typedef float v8f __attribute__((ext_vector_type(8)));

static __device__ __forceinline__ v8f wmma4(v2f a, v2f b, v8f c) {
  // V_WMMA_F32_16X16X4_F32: D = A(16x4) x B(4x16) + C(16x16)
  return __builtin_amdgcn_wmma_f32_16x16x4_f32(false, a, false, b, (short)0, c, false, false);
}
static __device__ __forceinline__ v8f bcast8(float x) {
  v8f r;
#pragma unroll
  for (int i = 0; i < 8; ++i) r[i] = x;
  return r;
}

// Async memory->LDS (GLOBAL_LOAD_ASYNC_TO_LDS_B128, GVS mode), tracked by ASYNCcnt.
// ldsOff: per-lane LDS byte offset (low 32 bits of a flat shared pointer)
// base:   uniform 64-bit global base (SGPR pair); off: per-lane signed byte offset
static __device__ __forceinline__ void async_ld128(unsigned ldsOff, const void* base, int off) {
#if defined(__gfx1250__)
  asm volatile("global_load_async_to_lds_b128 %0, %1, %2"
               :: "v"(ldsOff), "v"(off), "s"(base)
               : "memory");
#endif
}
static __device__ __forceinline__ void wait_async() {
#if defined(__gfx1250__)
  asm volatile("s_wait_asynccnt 0x0" ::: "memory");
#endif
}

// ---------------- FPS: one block per batch, serial farthest-point scan ----------------
__global__ void k_fps(const float* __restrict__ xyz, float* __restrict__ oxyz) {
  const int b = blockIdx.x;
  const int tid = threadIdx.x;  // 256 threads, 16 points each
  float px[16], py[16], pz[16], md[16];
#pragma unroll
  for (int j = 0; j < 16; ++j) {
    int n = tid + j * 256;
    const float* p = xyz + ((size_t)b * NN + n) * 3;
    px[j] = p[0]; py[j] = p[1]; pz[j] = p[2];
    md[j] = 1e10f;
  }
  __shared__ float cp[3];
  __shared__ float rv[8];
  __shared__ int ri[8];
  if (tid == 0) {
    const float* p0 = xyz + (size_t)b * NN * 3;
    cp[0] = p0[0]; cp[1] = p0[1]; cp[2] = p0[2];
    float* o = oxyz + (size_t)b * SS * 3;
    o[0] = p0[0]; o[1] = p0[1]; o[2] = p0[2];
  }
  __syncthreads();
  for (int step = 1; step < SS; ++step) {
    float bx = cp[0], by = cp[1], bz = cp[2];
    float bestv = -1.0f; int besti = 0x7fffffff;
#pragma unroll
    for (int j = 0; j < 16; ++j) {
      float dx = px[j] - bx, dy = py[j] - by, dz = pz[j] - bz;
      float d = dx * dx + dy * dy + dz * dz;
      md[j] = fminf(md[j], d);
      if (md[j] > bestv) { bestv = md[j]; besti = tid + j * 256; }
    }
#pragma unroll
    for (int mm = 16; mm >= 1; mm >>= 1) {
      float v2 = __shfl_xor(bestv, mm, 32);
      int i2 = __shfl_xor(besti, mm, 32);
      if (v2 > bestv || (v2 == bestv && i2 < besti)) { bestv = v2; besti = i2; }
    }
    if ((tid & 31) == 0) { rv[tid >> 5] = bestv; ri[tid >> 5] = besti; }
    __syncthreads();
    if (tid == 0) {
      float v = rv[0]; int ii = ri[0];
#pragma unroll
      for (int w = 1; w < 8; ++w)
        if (rv[w] > v || (rv[w] == v && ri[w] < ii)) { v = rv[w]; ii = ri[w]; }
      const float* p = xyz + ((size_t)b * NN + ii) * 3;
      cp[0] = p[0]; cp[1] = p[1]; cp[2] = p[2];
      float* o = oxyz + ((size_t)b * SS + step) * 3;
      o[0] = p[0]; o[1] = p[1]; o[2] = p[2];
    }
    __syncthreads();
  }
}

// ---------------- Ball query: one wave per query, ordered first-32 selection ----------------
__global__ void k_ballq(const float* __restrict__ xyz, const float* __restrict__ oxyz,
                        int* __restrict__ idxb) {
  int wave = blockIdx.x * 8 + (threadIdx.x >> 5);
  int lane = threadIdx.x & 31;
  int b = wave / SS;
  const float* q = oxyz + (size_t)wave * 3;
  float qx = q[0], qy = q[1], qz = q[2];
  int* out = idxb + (size_t)wave * NSK;
  int filled = 0;
  for (int base = 0; base < NN && filled < NSK; base += 32) {
    int n = base + lane;
    const float* p = xyz + ((size_t)b * NN + n) * 3;
    __builtin_prefetch(p + 96, 0, 0);  // next 32-point chunk
    float dx = p[0] - qx, dy = p[1] - qy, dz = p[2] - qz;
    float d = dx * dx + dy * dy + dz * dz;
    bool in = d < RAD2;
    unsigned mask = (unsigned)__ballot(in);
    int before = __popc(mask & ((1u << lane) - 1u));
    int slot = filled + before;
    if (in && slot < NSK) out[slot] = n;
    filled += __popc(mask);
  }
  int start = filled > NSK ? NSK : filled;
  for (int t = start + lane; t < NSK; t += 32) out[t] = -1;
}

// ---------------- BN1 stats: sum/sumsq of coc@w1_0+b1_0 over all B*S*NS*V rows ----------------
__global__ void k_stats1(const float* __restrict__ xyz, const float* __restrict__ fpnts,
                         const int* __restrict__ idxb, const float* __restrict__ w10,
                         const float* __restrict__ b10, float* __restrict__ s1) {
  int item = blockIdx.x * 256 + threadIdx.x;  // < 131072
  int b = item / (SS * NSK);
  int pi = idxb[item];
  float gx = 0, gy = 0, gz = 0;
  if (pi >= 0) {
    const float* p = xyz + ((size_t)b * NN + pi) * 3;
    gx = p[0]; gy = p[1]; gz = p[2];
  }
  float sacc[64], qacc[64];
#pragma unroll
  for (int c = 0; c < 64; ++c) { sacc[c] = 0; qacc[c] = 0; }
  for (int v = 0; v < VV; ++v) {
    float fx = fpnts[v * 3 + 0], fy = fpnts[v * 3 + 1], fz = fpnts[v * 3 + 2];
    float ax = gx - fx, ay = gy - fy, az = gz - fz;
    float eu = sqrtf(ax * ax + ay * ay + az * az);
    float coc[13] = {-ax, -ay, -az, ax, ay, az, eu, gx, gy, gz, fx, fy, fz};
#pragma unroll 4
    for (int c = 0; c < 64; ++c) {
      float h = b10[c];
#pragma unroll
      for (int k = 0; k < 13; ++k) h += coc[k] * w10[k * 64 + c];
      sacc[c] += h; qacc[c] += h * h;
    }
  }
  __shared__ float rs[64], rq[64];
  if (threadIdx.x < 64) { rs[threadIdx.x] = 0; rq[threadIdx.x] = 0; }
  __syncthreads();
#pragma unroll 4
  for (int c = 0; c < 64; ++c) { atomicAdd(&rs[c], sacc[c]); atomicAdd(&rq[c], qacc[c]); }
  __syncthreads();
  if (threadIdx.x < 64) {
    atomicAdd(&s1[threadIdx.x], rs[threadIdx.x]);
    atomicAdd(&s1[64 + threadIdx.x], rq[threadIdx.x]);
  }
}

// ---------------- Fold BN1 into layer-1 weights; pad w1_1 and w2_0 ----------------
__global__ void k_prep1(const float* __restrict__ s1, const float* __restrict__ w10,
                        const float* __restrict__ b10, const float* __restrict__ g1,
                        const float* __restrict__ bt1, const float* __restrict__ w11,
                        float* __restrict__ w1f, float* __restrict__ b1f,
                        float* __restrict__ w11p, const float* __restrict__ w20,
                        float* __restrict__ w20p) {
  int t = threadIdx.x;  // 256
  if (t < 64) {
    const float R = (float)((size_t)BB * SS * NSK * VV);
    float mean = s1[t] / R;
    float var = s1[64 + t] / R - mean * mean;
    float al = g1[t] * rsqrtf(var + EPSB);
    float be = bt1[t] - mean * al;
    b1f[t] = b10[t] * al + be;
#pragma unroll
    for (int k = 0; k < 16; ++k) w1f[k * 64 + t] = (k < 13) ? w10[k * 64 + t] * al : 0.0f;
  }
  for (int e = t; e < 64 * 80; e += 256) {
    int k = e / 80, n = e % 80;
    w11p[e] = (n < C1_) ? w11[k * C1_ + n] : 0.0f;
  }
  for (int e = t; e < 72 * 128; e += 256) {
    int k = e / 128;
    w20p[e] = (k < C1_) ? w20[e] : 0.0f;
  }
}

// ---------------- Main m1 pass: WMMA 13->64 (relu) -> 64->70, softmax over V, agg ----------------
__global__ void k_main1(const float* __restrict__ xyz, const float* __restrict__ f,
                        const float* __restrict__ fpnts, const int* __restrict__ idxb,
                        const float* __restrict__ oxyz, const float* __restrict__ w1f,
                        const float* __restrict__ b1f, const float* __restrict__ w11p,
                        const float* __restrict__ b11, float* __restrict__ agg) {
  __shared__ __attribute__((aligned(16))) float sm[8 * 1232];
  const int wid = threadIdx.x >> 5, lane = threadIdx.x & 31;
  float* hl = sm + wid * 1232;   // h bounce: 16 rows x stride 68
  float* gfl = hl + 1088;        // gf: 2 items x stride 72
  const int tile = blockIdx.x * 8 + wid;
  const int item0 = tile * 2;
  const int b = item0 / (SS * NSK);
  const int sr = item0 % (SS * NSK);
  const int s = sr / NSK;
  const int pidx0 = idxb[item0], pidx1 = idxb[item0 + 1];
  const float* q = oxyz + ((size_t)b * SS + s) * 3;
  const float qx = q[0], qy = q[1], qz = q[2];
  if (pidx0 >= 0) __builtin_prefetch(f + ((size_t)b * NN + pidx0) * CIN, 0, 0);
  if (pidx1 >= 0) __builtin_prefetch(f + ((size_t)b * NN + pidx1) * CIN, 0, 0);
  // stage gf = [gxyz(3), local(3), feat(64)] for the 2 items
  for (int e = lane; e < 2 * C1_; e += 32) {
    int il = (e < C1_) ? 0 : 1;
    int c = e - il * C1_;
    int pi = il ? pidx1 : pidx0;
    float val;
    if (c < 3) {
      val = (pi < 0) ? 0.0f : xyz[((size_t)b * NN + pi) * 3 + c];
    } else if (c < 6) {
      float g = (pi < 0) ? 0.0f : xyz[((size_t)b * NN + pi) * 3 + (c - 3)];
      float qc = (c == 3) ? qx : ((c == 4) ? qy : qz);
      val = g - qc;
    } else {
      val = (pi < 0) ? 0.0f : f[((size_t)b * NN + pi) * CIN + (c - 6)];
    }
    gfl[il * 72 + c] = val;
  }
  const int hi = lane >> 4;
  const int m = lane & 15;  // A-row / C-column-in-tile index
  {
    int il = (lane >> 3) & 1;
    int v = lane & 7;
    int pi = il ? pidx1 : pidx0;
    float gx = 0, gy = 0, gz = 0;
    if (pi >= 0) {
      const float* p = xyz + ((size_t)b * NN + pi) * 3;
      gx = p[0]; gy = p[1]; gz = p[2];
    }
    float fx = fpnts[v * 3 + 0], fy = fpnts[v * 3 + 1], fz = fpnts[v * 3 + 2];
    float ax = gx - fx, ay = gy - fy, az = gz - fz;
    float eu = sqrtf(ax * ax + ay * ay + az * az);
    float coc[16] = {-ax, -ay, -az, ax, ay, az, eu, gx, gy, gz, fx, fy, fz, 0.f, 0.f, 0.f};
    // layer 1: 13(pad16) -> 64, BN folded, C init = bias
    v8f a1[4];
#pragma unroll
    for (int nt = 0; nt < 4; ++nt) a1[nt] = bcast8(b1f[nt * 16 + m]);
#pragma unroll
    for (int s4 = 0; s4 < 4; ++s4) {
      v2f av;
      av.x = hi ? coc[4 * s4 + 2] : coc[4 * s4 + 0];
      av.y = hi ? coc[4 * s4 + 3] : coc[4 * s4 + 1];
      int kb = 4 * s4 + 2 * hi;
#pragma unroll
      for (int nt = 0; nt < 4; ++nt) {
        v2f bv;
        bv.x = w1f[kb * 64 + nt * 16 + m];
        bv.y = w1f[(kb + 1) * 64 + nt * 16 + m];
        a1[nt] = wmma4(av, bv, a1[nt]);
      }
    }
    // relu -> LDS (A-layout source for layer 2)
#pragma unroll
    for (int nt = 0; nt < 4; ++nt)
#pragma unroll
      for (int i = 0; i < 8; ++i)
        hl[(i + 8 * hi) * 68 + nt * 16 + m] = fmaxf(a1[nt][i], 0.0f);
  }
  // layer 2: 64 -> 70(pad80), C init = b1_1
  v8f a2[5];
#pragma unroll
  for (int nt = 0; nt < 5; ++nt) {
    int n2 = nt * 16 + m;
    a2[nt] = bcast8((n2 < C1_) ? b11[n2] : 0.0f);
  }
  for (int s4 = 0; s4 < 16; ++s4) {
    int kb = 4 * s4 + 2 * hi;
    v2f av = *reinterpret_cast<const v2f*>(&hl[m * 68 + kb]);
#pragma unroll
    for (int nt = 0; nt < 5; ++nt) {
      v2f bv;
      bv.x = w11p[kb * 80 + nt * 16 + m];
      bv.y = w11p[(kb + 1) * 80 + nt * 16 + m];
      a2[nt] = wmma4(av, bv, a2[nt]);
    }
  }
  // p = gf * a ; softmax over V axis == in-lane over the 8 C registers
  const bool mymask = (hi ? pidx1 : pidx0) < 0;
#pragma unroll
  for (int nt = 0; nt < 5; ++nt) {
    int c = nt * 16 + m;
    float g = (c < C1_) ? gfl[hi * 72 + c] : 0.0f;
    float p[8];
    float mx = -3.0e38f;
#pragma unroll
    for (int i = 0; i < 8; ++i) { p[i] = g * a2[nt][i]; mx = fmaxf(mx, p[i]); }
    float se = 0, sp = 0;
#pragma unroll
    for (int i = 0; i < 8; ++i) { float e = expf(p[i] - mx); se += e; sp += e * p[i]; }
    float av = sp / se;
    if (mymask) av = 0.0f;
    if (c < C1_) agg[(size_t)(item0 + hi) * C1_ + c] = av;
  }
}

// ---------------- BN2 stats via WMMA: sum/sumsq of agg@w2_0+b2_0 ----------------
__global__ void k_stats2(const float* __restrict__ agg, const float* __restrict__ w20p,
                         const float* __restrict__ b20, float* __restrict__ s2) {
  __shared__ __attribute__((aligned(16))) float sa[8 * 1216];
  __shared__ float rs[128], rq[128];
  const int wid = threadIdx.x >> 5, lane = threadIdx.x & 31;
  float* as = sa + wid * 1216;  // 16 rows x stride 76
  if (threadIdx.x < 128) { rs[threadIdx.x] = 0; rq[threadIdx.x] = 0; }
  __syncthreads();
  const int base = (blockIdx.x * 8 + wid) * 16;
  // async-stage 16 rows x 72 floats (18 x B128 chunks per row); cols 70..71 are
  // junk but multiply against zeroed w20p rows.
  {
    unsigned asBase = (unsigned)(uintptr_t)(void*)as;
    for (int e = lane; e < 16 * 18; e += 32) {
      int r = e / 18, ck = e % 18;
      unsigned ldsoff = asBase + (unsigned)((r * 76 + ck * 4) * 4);
      int goff = ((base + r) * C1_ + ck * 4) * 4;
      async_ld128(ldsoff, agg, goff);
    }
  }
  wait_async();
  const int hi = lane >> 4, m = lane & 15;
#pragma unroll
  for (int nt = 0; nt < 8; ++nt) {
    v8f acc = bcast8(b20[nt * 16 + m]);
    for (int s4 = 0; s4 < 18; ++s4) {
      int kb = 4 * s4 + 2 * hi;
      v2f av = *reinterpret_cast<const v2f*>(&as[m * 76 + kb]);
      v2f bv;
      bv.x = w20p[kb * 128 + nt * 16 + m];
      bv.y = w20p[(kb + 1) * 128 + nt * 16 + m];
      acc = wmma4(av, bv, acc);
    }
    float su = 0, sq = 0;
#pragma unroll
    for (int i = 0; i < 8; ++i) { su += acc[i]; sq += acc[i] * acc[i]; }
    atomicAdd(&rs[nt * 16 + m], su);
    atomicAdd(&rq[nt * 16 + m], sq);
  }
  __syncthreads();
  if (threadIdx.x < 128) {
    atomicAdd(&s2[threadIdx.x], rs[threadIdx.x]);
    atomicAdd(&s2[128 + threadIdx.x], rq[threadIdx.x]);
  }
}

// ---------------- Fold BN2 ----------------
__global__ void k_prep2(const float* __restrict__ s2, const float* __restrict__ w20,
                        const float* __restrict__ b20, const float* __restrict__ g2,
                        const float* __restrict__ bt2, float* __restrict__ w2f,
                        float* __restrict__ b2f) {
  int t = threadIdx.x;  // 128
  const float R = (float)((size_t)BB * SS * NSK);
  float mean = s2[t] / R;
  float var = s2[128 + t] / R - mean * mean;
  float al = g2[t] * rsqrtf(var + EPSB);
  float be = bt2[t] - mean * al;
  b2f[t] = b20[t] * al + be;
  for (int k = 0; k < 72; ++k) w2f[k * 128 + t] = (k < C1_) ? w20[k * 128 + t] * al : 0.0f;
}

// ---------------- Final: m2 (WMMA) + residual (WMMA into same C) + gelu + max over NS ----------------
__global__ void k_final(const float* __restrict__ f, const int* __restrict__ idxb,
                        const float* __restrict__ agg, const float* __restrict__ w2f,
                        const float* __restrict__ b2f, const float* __restrict__ w21,
                        const float* __restrict__ b21, const float* __restrict__ rw,
                        const float* __restrict__ rb, float* __restrict__ outF) {
  __shared__ __attribute__((aligned(16))) float sm[4 * 3200];
  const int wid = threadIdx.x >> 5, lane = threadIdx.x & 31;
  float* fs = sm + wid * 3200;  // feat: 16 x stride 68
  float* as = fs + 1088;        // agg: 16 x stride 76; reused as mid: 16 x stride 132
  float* midb = as;
  const int bs = blockIdx.x * 4 + wid;
  const int b = bs / SS;
  const int hi = lane >> 4, m = lane & 15;
  const unsigned fsBase = (unsigned)(uintptr_t)(void*)fs;
  const unsigned asBase = (unsigned)(uintptr_t)(void*)as;
  float vmax[8];
#pragma unroll
  for (int nt = 0; nt < 8; ++nt) vmax[nt] = -3.0e38f;
#pragma unroll 1
  for (int t = 0; t < 2; ++t) {
    const int base = bs * NSK + t * 16;
    // async-stage agg tile: 16 rows x 18 B128 chunks (cols 70..71 junk x zero rows)
    for (int e = lane; e < 16 * 18; e += 32) {
      int r = e / 18, ck = e % 18;
      unsigned ldsoff = asBase + (unsigned)((r * 76 + ck * 4) * 4);
      int goff = ((base + r) * C1_ + ck * 4) * 4;
      async_ld128(ldsoff, agg, goff);
    }
    // async-stage feat tile: 16 rows x 16 B128 chunks; masked rows pre-zeroed
    for (int e = lane; e < 16 * 16; e += 32) {
      int r = e >> 4, ck = e & 15;
      int pi = idxb[base + r];
      if (pi < 0) {
        v4f z = {0.f, 0.f, 0.f, 0.f};
        *reinterpret_cast<v4f*>(&fs[r * 68 + ck * 4]) = z;
      } else {
        unsigned ldsoff = fsBase + (unsigned)((r * 68 + ck * 4) * 4);
        int goff = (int)((((size_t)b * NN + pi) * CIN + ck * 4) * 4);
        async_ld128(ldsoff, f, goff);
      }
    }
    __builtin_prefetch(w21, 0, 0);
    wait_async();
    // L1: mid = relu(agg @ w2f + b2f)   (BN2 folded)
    v8f acc[8];
#pragma unroll
    for (int nt = 0; nt < 8; ++nt) acc[nt] = bcast8(b2f[nt * 16 + m]);
    for (int s4 = 0; s4 < 18; ++s4) {
      int kb = 4 * s4 + 2 * hi;
      v2f av = *reinterpret_cast<const v2f*>(&as[m * 76 + kb]);
#pragma unroll
      for (int nt = 0; nt < 8; ++nt) {
        v2f bv;
        bv.x = w2f[kb * 128 + nt * 16 + m];
        bv.y = w2f[(kb + 1) * 128 + nt * 16 + m];
        acc[nt] = wmma4(av, bv, acc[nt]);
      }
    }
#pragma unroll
    for (int nt = 0; nt < 8; ++nt)
#pragma unroll
      for (int i = 0; i < 8; ++i)
        midb[(i + 8 * hi) * 132 + nt * 16 + m] = fmaxf(acc[nt][i], 0.0f);
    // L2 + residual accumulate into one C
    v8f acc2[8];
#pragma unroll
    for (int nt = 0; nt < 8; ++nt) {
      int n = nt * 16 + m;
      acc2[nt] = bcast8(b21[n] + rb[n]);
    }
    for (int s4 = 0; s4 < 32; ++s4) {
      int kb = 4 * s4 + 2 * hi;
      v2f av = *reinterpret_cast<const v2f*>(&midb[m * 132 + kb]);
#pragma unroll
      for (int nt = 0; nt < 8; ++nt) {
        v2f bv;
        bv.x = w21[kb * 128 + nt * 16 + m];
        bv.y = w21[(kb + 1) * 128 + nt * 16 + m];
        acc2[nt] = wmma4(av, bv, acc2[nt]);
      }
    }
    for (int s4 = 0; s4 < 16; ++s4) {
      int kb = 4 * s4 + 2 * hi;
      v2f av = *reinterpret_cast<const v2f*>(&fs[m * 68 + kb]);
#pragma unroll
      for (int nt = 0; nt < 8; ++nt) {
        v2f bv;
        bv.x = rw[kb * 128 + nt * 16 + m];
        bv.y = rw[(kb + 1) * 128 + nt * 16 + m];
        acc2[nt] = wmma4(av, bv, acc2[nt]);
      }
    }
    // exact gelu + max over the 16 rows of this tile
#pragma unroll
    for (int nt = 0; nt < 8; ++nt) {
      float mx = -3.0e38f;
#pragma unroll
      for (int i = 0; i < 8; ++i) {
        float x = acc2[nt][i];
        float gl = 0.5f * x * (1.0f + erff(x * 0.70710678118654752f));
        mx = fmaxf(mx, gl);
      }
      float o = __shfl_xor(mx, 16, 32);
      mx = fmaxf(mx, o);
      vmax[nt] = fmaxf(vmax[nt], mx);
    }
  }
  if (lane < 16) {
#pragma unroll
    for (int nt = 0; nt < 8; ++nt)
      outF[(size_t)bs * COUT_ + nt * 16 + lane] = vmax[nt];
  }
}

extern "C" void kernel_launch(void* const* d_in, const int* in_sizes, int n_in,
                              void* d_out, int out_size, void* d_ws, size_t ws_size,
                              hipStream_t stream) {
  const float* xyz = (const float*)d_in[0];
  const float* f   = (const float*)d_in[1];
  const float* fp  = (const float*)d_in[2];
  const float* w10 = (const float*)d_in[3];
  const float* b10 = (const float*)d_in[4];
  const float* g1  = (const float*)d_in[5];
  const float* bt1 = (const float*)d_in[6];
  const float* w11 = (const float*)d_in[7];
  const float* b11 = (const float*)d_in[8];
  const float* w20 = (const float*)d_in[9];
  const float* b20 = (const float*)d_in[10];
  const float* g2  = (const float*)d_in[11];
  const float* bt2 = (const float*)d_in[12];
  const float* w21 = (const float*)d_in[13];
  const float* b21 = (const float*)d_in[14];
  const float* rw  = (const float*)d_in[15];
  const float* rb  = (const float*)d_in[16];

  float* outF = (float*)d_out;
  float* outX = outF + (size_t)BB * SS * COUT_;  // new_xyz output region

  // workspace layout (floats)
  int*   idxb = (int*)d_ws;                        // 131072 ints
  float* wsf  = (float*)d_ws;
  float* agg  = wsf + 131072;                      // 131072 x 70
  float* s1   = agg + (size_t)131072 * C1_;        // 128
  float* s2   = s1 + 128;                          // 256
  float* w1f  = s2 + 256;                          // 16 x 64 (BN1-folded, K-padded)
  float* b1f  = w1f + 16 * 64;                     // 64
  float* w11p = b1f + 64;                          // 64 x 80 (N-padded)
  float* w20p = w11p + 64 * 80;                    // 72 x 128 (K-padded)
  float* w2f  = w20p + 72 * 128;                   // 72 x 128 (BN2-folded, K-padded)
  float* b2f  = w2f + 72 * 128;                    // 128

  hipMemsetAsync(s1, 0, (128 + 256) * sizeof(float), stream);
  k_fps<<<BB, 256, 0, stream>>>(xyz, outX);
  k_ballq<<<512, 256, 0, stream>>>(xyz, outX, idxb);
  k_stats1<<<512, 256, 0, stream>>>(xyz, fp, idxb, w10, b10, s1);
  k_prep1<<<1, 256, 0, stream>>>(s1, w10, b10, g1, bt1, w11, w1f, b1f, w11p, w20, w20p);
  k_main1<<<8192, 256, 0, stream>>>(xyz, f, fp, idxb, outX, w1f, b1f, w11p, b11, agg);
  k_stats2<<<1024, 256, 0, stream>>>(agg, w20p, b20, s2);
  k_prep2<<<1, 128, 0, stream>>>(s2, w20, b20, g2, bt2, w2f, b2f);
  k_final<<<1024, 128, 0, stream>>>(f, idxb, agg, w2f, b2f, w21, b21, rw, rb, outF);
}